// SetPredLoss_39075612459658
// MI455X (gfx1250) — compile-verified
//
#include <hip/hip_runtime.h>
#include <hip/hip_bf16.h>

typedef __attribute__((ext_vector_type(2))) float v2f;
typedef __attribute__((ext_vector_type(8))) float v8f;

#define NS 16
#define DIM 64

// One wave32 workgroup per batch element:
//   1) stage x/t tiles (16x64 f32 each) in LDS (padded rows -> no bank conflicts)
//   2) row norms
//   3) Gram G = x . t^T via 16x V_WMMA_F32_16X16X4_F32 (K=64), full f32
//   4) huber correction  corr = sum_d 0.5*max(|x-t|-1,0)^2  (exact decomposition:
//      smooth_l1(d) = 0.5 d^2 - 0.5 max(|d|-1,0)^2)
//   5) cost[i][j] = (0.5(|x_i|^2+|t_j|^2) - G[i][j] - corr[i][j]) / 64
//   6) lane-parallel Jonker-Volgenant Hungarian (columns across lanes, shuffles)
//   7) matched-cost wave sum -> per_batch[b]
__global__ __launch_bounds__(32) void setpred_cost_hungarian(
    const float* __restrict__ x, const float* __restrict__ t,
    float* __restrict__ per_batch, int B)
{
    __shared__ float xs[NS][DIM + 1];
    __shared__ float ts[NS][DIM + 1];
    __shared__ float cost[NS][NS + 1];
    __shared__ float nrmx[NS];
    __shared__ float nrmt[NS];

    const int b = blockIdx.x;
    if (b >= B) return;
    const int lane = threadIdx.x;   // 0..31
    const int lo = lane & 15;
    const int hi = lane >> 4;

    // ---- stage tiles, coalesced b128 loads ----
    const float4* xg = (const float4*)(x + (size_t)b * NS * DIM);
    const float4* tg = (const float4*)(t + (size_t)b * NS * DIM);
    #pragma unroll
    for (int idx = lane; idx < NS * (DIM / 4); idx += 32) {
        const int row = idx >> 4;       // 16 float4 chunks per row
        const int c4  = (idx & 15) * 4;
        float4 vx = xg[idx];
        float4 vt = tg[idx];
        xs[row][c4 + 0] = vx.x; xs[row][c4 + 1] = vx.y;
        xs[row][c4 + 2] = vx.z; xs[row][c4 + 3] = vx.w;
        ts[row][c4 + 0] = vt.x; ts[row][c4 + 1] = vt.y;
        ts[row][c4 + 2] = vt.z; ts[row][c4 + 3] = vt.w;
    }
    __syncthreads();

    // ---- row norms: lanes 0-15 -> x rows, lanes 16-31 -> t rows ----
    {
        const float* row = (hi == 0) ? xs[lo] : ts[lo];
        float s = 0.f;
        #pragma unroll 8
        for (int dd = 0; dd < DIM; ++dd) s += row[dd] * row[dd];
        if (hi == 0) nrmx[lo] = s; else nrmt[lo] = s;
    }

    // ---- Gram G = x . t^T via V_WMMA_F32_16X16X4_F32, accumulate K=64 ----
    // A 16x4 layout: lane m (both halves), VGPR0 = K {0|2}, VGPR1 = K {1|3}
    // B 4x16 layout: lane n (both halves), same K striping; B[k][n] = t[n][k]
    v8f acc = {};
    const int kb = hi * 2;
    #pragma unroll
    for (int kk = 0; kk < DIM; kk += 4) {
        v2f a, bm;
        a[0]  = xs[lo][kk + kb];
        a[1]  = xs[lo][kk + kb + 1];
        bm[0] = ts[lo][kk + kb];
        bm[1] = ts[lo][kk + kb + 1];
        acc = __builtin_amdgcn_wmma_f32_16x16x4_f32(
            /*neg_a=*/false, a, /*neg_b=*/false, bm,
            /*c_mod=*/(short)0, acc, /*reuse_a=*/false, /*reuse_b=*/false);
    }

    __syncthreads();  // norms visible to all lanes

    // ---- huber correction + combine; D layout: VGPR r -> row r + 8*hi, col lo ----
    #pragma unroll
    for (int r = 0; r < 8; ++r) {
        const int i = r + 8 * hi;
        const int j = lo;
        const float* xr = xs[i];
        const float* tr = ts[j];
        float c = 0.f;
        #pragma unroll 8
        for (int dd = 0; dd < DIM; ++dd) {
            float df = xr[dd] - tr[dd];
            float e  = fmaxf(fabsf(df) - 1.0f, 0.0f);
            c += 0.5f * e * e;
        }
        cost[i][j] = (0.5f * (nrmx[i] + nrmt[j]) - acc[r] - c) * (1.0f / 64.0f);
    }
    __syncthreads();

    // ---- lane-parallel Jonker-Volgenant ----
    const float INF = __builtin_inff();
    float u_val = 0.f;   // lane r holds u[r+1]
    float v_val = 0.f;   // lane j holds v[j]
    int   p_val = 0;     // lane j holds p[j] (row matched to column j)

    for (int i = 1; i <= NS; ++i) {
        if (lane == 0) p_val = i;   // p[0] = i
        int   j0 = 0;
        float minv = INF;
        int   way = 0;
        bool  used = false;
        bool  rowused = false;
        for (;;) {
            if (lane == j0) used = true;
            const int i0 = __shfl(p_val, j0);
            if (lane == i0 - 1) rowused = true;
            const float u_i0 = __shfl(u_val, i0 - 1);
            const bool active = (lane >= 1) && (lane <= NS) && !used;
            if (active) {
                float cur = cost[i0 - 1][lane - 1] - u_i0 - v_val;
                if (cur < minv) { minv = cur; way = j0; }
            }
            // argmin over active lanes (lowest index on ties, like np.argmin)
            float bv = active ? minv : INF;
            int   bj = lane;
            #pragma unroll
            for (int off = 16; off; off >>= 1) {
                float ov = __shfl_xor(bv, off);
                int   oj = __shfl_xor(bj, off);
                if (ov < bv || (ov == bv && oj < bj)) { bv = ov; bj = oj; }
            }
            const float delta = bv;
            if (rowused) u_val += delta;
            if (used) v_val -= delta; else minv -= delta;
            j0 = bj;
            const int pj0 = __shfl(p_val, j0);
            if (pj0 == 0) break;
        }
        // augment along the alternating path
        for (;;) {
            const int j1  = __shfl(way, j0);
            const int pj1 = __shfl(p_val, j1);
            if (lane == j0) p_val = pj1;
            j0 = j1;
            if (j0 == 0) break;
        }
    }

    // ---- matched cost sum: column j matched to row p[j] ----
    float mval = 0.f;
    if (lane >= 1 && lane <= NS) mval = cost[p_val - 1][lane - 1];
    #pragma unroll
    for (int off = 16; off; off >>= 1) mval += __shfl_xor(mval, off);
    if (lane == 0) per_batch[b] = mval;
}

__global__ __launch_bounds__(256) void reduce_mean(
    const float* __restrict__ pb, float* __restrict__ out, int B)
{
    __shared__ float sh[256];
    float s = 0.f;
    for (int i = threadIdx.x; i < B; i += 256) s += pb[i];
    sh[threadIdx.x] = s;
    __syncthreads();
    #pragma unroll
    for (int stride = 128; stride > 0; stride >>= 1) {
        if (threadIdx.x < stride) sh[threadIdx.x] += sh[threadIdx.x + stride];
        __syncthreads();
    }
    if (threadIdx.x == 0) out[0] = sh[0] / (float)B;
}

extern "C" void kernel_launch(void* const* d_in, const int* in_sizes, int n_in,
                              void* d_out, int out_size, void* d_ws, size_t ws_size,
                              hipStream_t stream) {
    const float* x = (const float*)d_in[0];
    const float* t = (const float*)d_in[1];
    float* out = (float*)d_out;
    float* pb  = (float*)d_ws;                 // B floats of scratch
    const int B = in_sizes[0] / (NS * DIM);    // 4096 for the reference shapes

    setpred_cost_hungarian<<<dim3(B), dim3(32), 0, stream>>>(x, t, pb, B);
    reduce_mean<<<dim3(1), dim3(256), 0, stream>>>(pb, out, B);
}